// get_model_1340029796524
// MI455X (gfx1250) — compile-verified
//
#include <hip/hip_runtime.h>
#include <hip/hip_bf16.h>

// ---------------------------------------------------------------------------
// PointNet++ (SSG) forward for MI455X / gfx1250.
// fp32 end-to-end; matmuls on the matrix pipe via V_WMMA_F32_16X16X4_F32.
// ---------------------------------------------------------------------------

typedef float v2f __attribute__((ext_vector_type(2)));
typedef float v8f __attribute__((ext_vector_type(8)));

#define BATCH 32
#define NPTS  4096
#define EPSBN 1e-5f

// ---------------------------------------------------------------------------
// Farthest point sampling: one block (256 threads) per batch element.
// Source is strided: src[b*sb + n*sn + d*sd], d in {0,1,2}.
// Records `farthest` BEFORE the update each step (matches jax.lax.scan output).
// ---------------------------------------------------------------------------
__global__ void k_fps(const float* __restrict__ src, long sb, long sn, long sd,
                      int* __restrict__ fidx, int N, int S) {
  const int b = blockIdx.x;
  const int t = threadIdx.x;
  const int T = 256;
  const float* p = src + (long)b * sb;
  const int per = N / T;          // 16 for N=4096, 2 for N=512
  float dist[16];
  for (int i = 0; i < per; ++i) dist[i] = 1e10f;

  __shared__ float s_val[256];
  __shared__ int   s_idx[256];

  int far = 0;
  for (int s = 0; s < S; ++s) {
    if (t == 0) fidx[b * S + s] = far;
    const float cx = p[(long)far * sn + 0 * sd];
    const float cy = p[(long)far * sn + 1 * sd];
    const float cz = p[(long)far * sn + 2 * sd];
    float bestv = -1.0f; int besti = N;
    for (int i = 0; i < per; ++i) {
      const int n = t + i * T;    // coalesced
      const float dx = p[(long)n * sn + 0 * sd] - cx;
      const float dy = p[(long)n * sn + 1 * sd] - cy;
      const float dz = p[(long)n * sn + 2 * sd] - cz;
      const float d  = dx * dx + dy * dy + dz * dz;
      const float dm = fminf(dist[i], d);
      dist[i] = dm;
      if (dm > bestv || (dm == bestv && n < besti)) { bestv = dm; besti = n; }
    }
    s_val[t] = bestv; s_idx[t] = besti;
    __syncthreads();
    for (int off = 128; off > 0; off >>= 1) {
      if (t < off) {
        const float v2 = s_val[t + off]; const int i2 = s_idx[t + off];
        if (v2 > s_val[t] || (v2 == s_val[t] && i2 < s_idx[t])) {
          s_val[t] = v2; s_idx[t] = i2;
        }
      }
      __syncthreads();
    }
    far = s_idx[0];
    __syncthreads();
  }
}

// Gather sampled centers into (B,S,3) row-major.
__global__ void k_centers(const float* __restrict__ src, long sb, long sn, long sd,
                          const int* __restrict__ fidx, float* __restrict__ nxyz,
                          int S) {
  const int i = blockIdx.x * blockDim.x + threadIdx.x;   // over B*S
  if (i >= BATCH * S) return;
  const int b = i / S;
  const int n = fidx[i];
  const float* p = src + (long)b * sb + (long)n * sn;
  nxyz[i * 3 + 0] = p[0 * sd];
  nxyz[i * 3 + 1] = p[1 * sd];
  nxyz[i * 3 + 2] = p[2 * sd];
}

// Ball query: index-ordered scan collecting first K in-radius points; this is
// exactly the reference's mask->sort->slice->fill-with-first scheme.
// One block per batch, one thread per center (S <= 512).
__global__ void k_ballquery(const float* __restrict__ src, long sb, long sn, long sd,
                            const float* __restrict__ nxyz, int* __restrict__ gidx,
                            int N, int S, int K, float r2) {
  const int b = blockIdx.x;
  const int s = threadIdx.x;
  if (s >= S) return;
  const long bs = (long)b * S + s;
  const float cx = nxyz[bs * 3 + 0];
  const float cy = nxyz[bs * 3 + 1];
  const float cz = nxyz[bs * 3 + 2];
  int* g = gidx + bs * K;
  int cnt = 0, first = 0;
  const float* p = src + (long)b * sb;
  for (int n = 0; n < N && cnt < K; ++n) {
    const float dx = p[(long)n * sn + 0 * sd] - cx;
    const float dy = p[(long)n * sn + 1 * sd] - cy;
    const float dz = p[(long)n * sn + 2 * sd] - cz;
    if (dx * dx + dy * dy + dz * dz <= r2) {
      if (cnt == 0) first = n;
      g[cnt++] = n;
    }
  }
  for (; cnt < K; ++cnt) g[cnt] = first;
}

// SA1 grouping: xyz is (B,6,N) strided. feat row (b,s,k) = [gxyz(3), norm(3)].
__global__ void k_group_sa1(const float* __restrict__ xyz,
                            const float* __restrict__ nxyz,
                            const int* __restrict__ gidx,
                            float* __restrict__ feat, int S, int K) {
  const long id = (long)blockIdx.x * blockDim.x + threadIdx.x;  // B*S*K
  const long total = (long)BATCH * S * K;
  if (id >= total) return;
  const long bs = id / K;
  const int  b  = (int)(bs / S);
  const int  n  = gidx[id];
  const float* p = xyz + (long)b * 6 * NPTS;
  float* f = feat + id * 6;
  f[0] = p[0 * NPTS + n] - nxyz[bs * 3 + 0];
  f[1] = p[1 * NPTS + n] - nxyz[bs * 3 + 1];
  f[2] = p[2 * NPTS + n] - nxyz[bs * 3 + 2];
  f[3] = p[3 * NPTS + n];
  f[4] = p[4 * NPTS + n];
  f[5] = p[5 * NPTS + n];
}

// SA2 grouping: prev points (B,Np,3)+(B,Np,Cp) row-major; feat = [gxyz, pts].
__global__ void k_group_sa2(const float* __restrict__ px,   // (B,Np,3)
                            const float* __restrict__ pf,   // (B,Np,Cp)
                            const float* __restrict__ nxyz,
                            const int* __restrict__ gidx,
                            float* __restrict__ feat,
                            int Np, int S, int K, int Cp) {
  const long id = (long)blockIdx.x * blockDim.x + threadIdx.x;  // B*S*K
  const long total = (long)BATCH * S * K;
  if (id >= total) return;
  const long bs = id / K;
  const int  b  = (int)(bs / S);
  const int  n  = gidx[id];
  const int  C  = 3 + Cp;
  float* f = feat + id * C;
  const float* q = px + ((long)b * Np + n) * 3;
  f[0] = q[0] - nxyz[bs * 3 + 0];
  f[1] = q[1] - nxyz[bs * 3 + 1];
  f[2] = q[2] - nxyz[bs * 3 + 2];
  const float* r = pf + ((long)b * Np + n) * Cp;
  for (int c = 0; c < Cp; ++c) f[3 + c] = r[c];
}

// SA3 group_all: feat row (b,k) = [l2x(3), l2p(256)]  (no centering).
__global__ void k_group_sa3(const float* __restrict__ px,   // (B,128,3)
                            const float* __restrict__ pf,   // (B,128,256)
                            float* __restrict__ feat) {
  const int id = blockIdx.x * blockDim.x + threadIdx.x;     // B*128
  if (id >= BATCH * 128) return;
  float* f = feat + (long)id * 259;
  const float* q = px + (long)id * 3;
  f[0] = q[0]; f[1] = q[1]; f[2] = q[2];
  const float* r = pf + (long)id * 256;
  for (int c = 0; c < 256; ++c) f[3 + c] = r[c];
}

// ---------------------------------------------------------------------------
// GEMM: Y[r,o] = sum_c X[r,c] * W[o,c] + bias[o]
// X (R,C) row-major, W (O,C) row-major, R multiple of 16 (true for all calls).
// One wave per 16x16 output tile; K-loop in steps of 4 using fp32 WMMA.
//
// A layout (16x4 f32): lanes 0-15 -> row M=lane, {k,k+1}; lanes 16-31 -> row
// M=lane-16, {k+2,k+3}.  B tile mirrors with N across lanes.  D: VGPR i holds
// M = i + 8*(lane>=16), N = lane&15.
//
// The inner loop is fully branch-free: the multiple-of-4 body issues
// unconditional loads (all 16 rows valid by construction; OOB weight columns
// are clamped to row 0 of W — they only affect output columns we never
// store).  The <=1 remainder iteration uses address clamp + select (cndmask),
// not conditional loads, so EXEC never changes inside the kernel (WMMA
// requires EXEC all-1s).
// ---------------------------------------------------------------------------
__global__ void k_gemm_bias(const float* __restrict__ X, const float* __restrict__ W,
                            const float* __restrict__ bias, float* __restrict__ Y,
                            int R, int C, int O) {
  const int OT    = (O + 15) >> 4;
  const long wave = ((long)blockIdx.x * blockDim.x + threadIdx.x) >> 5;
  const int lane  = threadIdx.x & 31;
  const long tilesR = R >> 4;
  if (wave >= tilesR * OT) return;           // wave-uniform exit
  const int tileO  = (int)(wave % OT);
  const long tileR = wave / OT;
  const int rowBase = (int)(tileR << 4);
  const int colBase = tileO << 4;

  const int half = lane >> 4;                // 0 -> {k,k+1}, 1 -> {k+2,k+3}
  const int lrow = lane & 15;
  const int col  = colBase + lrow;
  const bool colOK = (col < O);

  // Per-lane base pointers, including the +2*half K offset.
  const float* xr = X + (long)(rowBase + lrow) * C + 2 * half;
  const float* wr = W + (long)(colOK ? col : 0) * C + 2 * half;

  const int Cm = C & ~3;                     // main (multiple-of-4) extent
  v8f acc = {};
  int k0 = 0;
  for (; k0 < Cm; k0 += 4) {                 // branch-free steady state
    v2f af, bf;
    af.x = xr[k0]; af.y = xr[k0 + 1];
    bf.x = wr[k0]; bf.y = wr[k0 + 1];
    acc = __builtin_amdgcn_wmma_f32_16x16x4_f32(
        false, af, false, bf, (short)0, acc, false, false);
  }
  if (Cm < C) {                              // <=1 zero-padded tail iteration
    const int ra = k0 + 2 * half;            // logical k of .x element
    // address-clamped loads + selects: no EXEC divergence
    const float ax = xr[(ra     < C) ? k0     : -2 * half];
    const float ay = xr[(ra + 1 < C) ? k0 + 1 : -2 * half];
    const float bx = wr[(ra     < C) ? k0     : -2 * half];
    const float by = wr[(ra + 1 < C) ? k0 + 1 : -2 * half];
    v2f af, bf;
    af.x = (ra     < C) ? ax : 0.0f;
    af.y = (ra + 1 < C) ? ay : 0.0f;
    bf.x = (ra     < C) ? bx : 0.0f;
    bf.y = (ra + 1 < C) ? by : 0.0f;
    acc = __builtin_amdgcn_wmma_f32_16x16x4_f32(
        false, af, false, bf, (short)0, acc, false, false);
  }

  if (colOK) {
    const float bb = bias[col];
#pragma unroll
    for (int i = 0; i < 8; ++i) {
      const int m = i + 8 * half;
      Y[(long)(rowBase + m) * O + col] = acc[i] + bb;
    }
  }
}

// ---------------------------------------------------------------------------
// BatchNorm with live batch statistics over all rows (axes (0,1,2) / (0,)):
// pass 1: per-channel sum & sum-of-squares (block partials + f32 atomics),
// pass 2: normalize + affine + ReLU in place.
// ---------------------------------------------------------------------------
__global__ void k_zero2(float* a, float* b, int n) {
  const int i = blockIdx.x * blockDim.x + threadIdx.x;
  if (i < n) { a[i] = 0.0f; b[i] = 0.0f; }
}

__global__ void k_colstats(const float* __restrict__ Y, float* __restrict__ sum,
                           float* __restrict__ sumsq, int R, int O, int rpb) {
  const int c = blockIdx.y * blockDim.x + threadIdx.x;
  if (c >= O) return;
  const int r0 = blockIdx.x * rpb;
  const int r1 = min(R, r0 + rpb);
  float s = 0.0f, q = 0.0f;
  for (int r = r0; r < r1; ++r) {
    const float v = Y[(long)r * O + c];     // coalesced across threads
    s += v; q += v * v;
  }
  atomicAdd(sum + c, s);
  atomicAdd(sumsq + c, q);
}

__global__ void k_bn_relu(float* __restrict__ Y, const float* __restrict__ sum,
                          const float* __restrict__ sumsq,
                          const float* __restrict__ g, const float* __restrict__ bt,
                          long R, int O) {
  const long i = (long)blockIdx.x * blockDim.x + threadIdx.x;
  if (i >= R * O) return;
  const int c = (int)(i % O);
  const float inv = 1.0f / (float)R;
  const float m = sum[c] * inv;
  const float v = sumsq[c] * inv - m * m;
  const float y = (Y[i] - m) * rsqrtf(v + EPSBN) * g[c] + bt[c];
  Y[i] = fmaxf(0.0f, y);
}

// Max over the neighbor axis: out[bs,c] = max_k feat[(bs*K+k)*O + c].
__global__ void k_maxpool(const float* __restrict__ feat, float* __restrict__ out,
                          long BS, int K, int O) {
  const long i = (long)blockIdx.x * blockDim.x + threadIdx.x;
  if (i >= BS * O) return;
  const int  c  = (int)(i % O);
  const long bs = i / O;
  const float* f = feat + bs * (long)K * O + c;
  float m = f[0];
  for (int k = 1; k < K; ++k) m = fmaxf(m, f[(long)k * O]);
  out[i] = m;
}

__global__ void k_logsoftmax(const float* __restrict__ h, float* __restrict__ out,
                             int B, int O) {
  const int b = blockIdx.x * blockDim.x + threadIdx.x;
  if (b >= B) return;
  const float* r = h + (long)b * O;
  float mx = r[0];
  for (int c = 1; c < O; ++c) mx = fmaxf(mx, r[c]);
  float s = 0.0f;
  for (int c = 0; c < O; ++c) s += __expf(r[c] - mx);
  const float l = __logf(s) + mx;
  float* o = out + (long)b * O;
  for (int c = 0; c < O; ++c) o[c] = r[c] - l;
}

// ---------------------------------------------------------------------------
// Host-side orchestration
// ---------------------------------------------------------------------------
static inline void launch_gemm(const float* X, const float* W, const float* b,
                               float* Y, int R, int C, int O, hipStream_t st) {
  const int OT = (O + 15) / 16;
  const long waves = (long)(R / 16) * OT;
  const long threads = waves * 32;
  const unsigned blocks = (unsigned)((threads + 255) / 256);
  k_gemm_bias<<<blocks, 256, 0, st>>>(X, W, b, Y, R, C, O);
}

static inline void launch_bn_relu(float* Y, long R, int O, const float* g,
                                  const float* bt, float* sum, float* sq,
                                  hipStream_t st) {
  k_zero2<<<(O + 255) / 256, 256, 0, st>>>(sum, sq, O);
  const int rpb = 2048;
  dim3 gs((unsigned)((R + rpb - 1) / rpb), (unsigned)((O + 255) / 256));
  k_colstats<<<gs, 256, 0, st>>>(Y, sum, sq, (int)R, O, rpb);
  const long n = R * O;
  k_bn_relu<<<(unsigned)((n + 255) / 256), 256, 0, st>>>(Y, sum, sq, g, bt, R, O);
}

extern "C" void kernel_launch(void* const* d_in, const int* in_sizes, int n_in,
                              void* d_out, int out_size, void* d_ws, size_t ws_size,
                              hipStream_t stream) {
  (void)in_sizes; (void)n_in; (void)out_size; (void)ws_size;
  // Inputs in setup_inputs() dict order: xyz, then params (insertion order):
  //  sa1: 3x(W,b,g,bt)  -> idx 1..12
  //  sa2: 3x(W,b,g,bt)  -> idx 13..24
  //  sa3: 3x(W,b,g,bt)  -> idx 25..36
  //  fc1 (W,b)=37,38  bn1 (g,bt)=39,40  fc2=41,42  bn2=43,44  fc3=45,46
  const float* xyz = (const float*)d_in[0];           // (32, 6, 4096)
  auto P = [&](int i) { return (const float*)d_in[i]; };

  // Workspace carve-up (256B aligned).
  char* w = (char*)d_ws;
  auto carve = [&](size_t bytes) -> void* {
    void* p = (void*)w;
    w += (bytes + 255) & ~(size_t)255;
    return p;
  };
  float* featA = (float*)carve((size_t)BATCH * 512 * 32 * 128 * 4);  // 268 MB
  float* featB = (float*)carve((size_t)BATCH * 512 * 32 * 128 * 4);  // 268 MB
  float* l1p   = (float*)carve((size_t)BATCH * 512 * 128 * 4);
  float* l2p   = (float*)carve((size_t)BATCH * 128 * 256 * 4);
  float* nx1   = (float*)carve((size_t)BATCH * 512 * 3 * 4);         // l1x
  float* nx2   = (float*)carve((size_t)BATCH * 128 * 3 * 4);         // l2x
  int*   fps   = (int*)carve((size_t)BATCH * 512 * 4);
  int*   gidx  = (int*)carve((size_t)BATCH * 512 * 32 * 4);
  float* sum   = (float*)carve(1024 * 4);
  float* sq    = (float*)carve(1024 * 4);
  float* x1    = (float*)carve((size_t)BATCH * 512 * 4);
  float* x2    = (float*)carve((size_t)BATCH * 256 * 4);
  float* h3    = (float*)carve((size_t)BATCH * 40 * 4);

  float* logits = (float*)d_out;                 // (32,40)
  float* l3p    = (float*)d_out + BATCH * 40;    // (32,1,1024)

  // ---------------- SA1: N=4096 -> S=512, K=32, C 6->64->64->128 -----------
  {
    const int S = 512, K = 32;
    const long sb = 6L * NPTS, sn = 1, sd = NPTS;  // xyz (B,6,N) strided
    k_fps<<<BATCH, 256, 0, stream>>>(xyz, sb, sn, sd, fps, NPTS, S);
    k_centers<<<(BATCH * S + 255) / 256, 256, 0, stream>>>(xyz, sb, sn, sd, fps, nx1, S);
    k_ballquery<<<BATCH, S, 0, stream>>>(xyz, sb, sn, sd, nx1, gidx, NPTS, S, K, 0.2f * 0.2f);
    const long R = (long)BATCH * S * K;            // 524288
    k_group_sa1<<<(unsigned)((R + 255) / 256), 256, 0, stream>>>(xyz, nx1, gidx, featA, S, K);
    launch_gemm(featA, P(1), P(2), featB, (int)R, 6, 64, stream);
    launch_bn_relu(featB, R, 64, P(3), P(4), sum, sq, stream);
    launch_gemm(featB, P(5), P(6), featA, (int)R, 64, 64, stream);
    launch_bn_relu(featA, R, 64, P(7), P(8), sum, sq, stream);
    launch_gemm(featA, P(9), P(10), featB, (int)R, 64, 128, stream);
    launch_bn_relu(featB, R, 128, P(11), P(12), sum, sq, stream);
    const long BS = (long)BATCH * S;
    k_maxpool<<<(unsigned)((BS * 128 + 255) / 256), 256, 0, stream>>>(featB, l1p, BS, K, 128);
  }

  // ---------------- SA2: N=512 -> S=128, K=64, C 131->128->128->256 --------
  {
    const int Np = 512, S = 128, K = 64;
    const long sb = (long)Np * 3, sn = 3, sd = 1;  // nx1 (B,512,3) row-major
    k_fps<<<BATCH, 256, 0, stream>>>(nx1, sb, sn, sd, fps, Np, S);
    k_centers<<<(BATCH * S + 255) / 256, 256, 0, stream>>>(nx1, sb, sn, sd, fps, nx2, S);
    k_ballquery<<<BATCH, S, 0, stream>>>(nx1, sb, sn, sd, nx2, gidx, Np, S, K, 0.4f * 0.4f);
    const long R = (long)BATCH * S * K;            // 262144
    k_group_sa2<<<(unsigned)((R + 255) / 256), 256, 0, stream>>>(nx1, l1p, nx2, gidx, featA, Np, S, K, 128);
    launch_gemm(featA, P(13), P(14), featB, (int)R, 131, 128, stream);
    launch_bn_relu(featB, R, 128, P(15), P(16), sum, sq, stream);
    launch_gemm(featB, P(17), P(18), featA, (int)R, 128, 128, stream);
    launch_bn_relu(featA, R, 128, P(19), P(20), sum, sq, stream);
    launch_gemm(featA, P(21), P(22), featB, (int)R, 128, 256, stream);
    launch_bn_relu(featB, R, 256, P(23), P(24), sum, sq, stream);
    const long BS = (long)BATCH * S;
    k_maxpool<<<(unsigned)((BS * 256 + 255) / 256), 256, 0, stream>>>(featB, l2p, BS, K, 256);
  }

  // ---------------- SA3: group_all, K=128, C 259->256->512->1024 -----------
  {
    const long R = (long)BATCH * 128;              // 4096
    k_group_sa3<<<(unsigned)((R + 255) / 256), 256, 0, stream>>>(nx2, l2p, featA);
    launch_gemm(featA, P(25), P(26), featB, (int)R, 259, 256, stream);
    launch_bn_relu(featB, R, 256, P(27), P(28), sum, sq, stream);
    launch_gemm(featB, P(29), P(30), featA, (int)R, 256, 512, stream);
    launch_bn_relu(featA, R, 512, P(31), P(32), sum, sq, stream);
    launch_gemm(featA, P(33), P(34), featB, (int)R, 512, 1024, stream);
    launch_bn_relu(featB, R, 1024, P(35), P(36), sum, sq, stream);
    k_maxpool<<<(unsigned)(((long)BATCH * 1024 + 255) / 256), 256, 0, stream>>>(featB, l3p, BATCH, 128, 1024);
  }

  // ---------------- FC head ------------------------------------------------
  launch_gemm(l3p, P(37), P(38), x1, BATCH, 1024, 512, stream);
  launch_bn_relu(x1, BATCH, 512, P(39), P(40), sum, sq, stream);
  launch_gemm(x1, P(41), P(42), x2, BATCH, 512, 256, stream);
  launch_bn_relu(x2, BATCH, 256, P(43), P(44), sum, sq, stream);
  launch_gemm(x2, P(45), P(46), h3, BATCH, 256, 40, stream);
  k_logsoftmax<<<1, BATCH, 0, stream>>>(h3, logits, BATCH, 40);
}